// Model_64690797412753
// MI455X (gfx1250) — compile-verified
//
#include <hip/hip_runtime.h>
#include <math.h>

// Problem dims (from reference)
#define Bb   16
#define Gg   30
#define Hd   24
#define Wd   862
#define PREQ 720          // G*H
#define HW   (Hd*Wd)      // 20688
#define NR   2
#define MTW  15           // 720 / 48 : M-supertiles of 48 rows (3 WMMA tiles)
#define NTW  27           // ceil(862/32): N-supertiles of 32 cols (2 WMMA tiles)

typedef __attribute__((ext_vector_type(2))) float v2f;
typedef __attribute__((ext_vector_type(8))) float v8f;

// ---------------------------------------------------------------------------
// Kernel 1: fused group_inner + 2 deformable-conv branches + softmax gate
// one thread per pixel (b,h,w), produces repres [B,G,H,W]
// ---------------------------------------------------------------------------
__global__ void fused_gate_kernel(const float* __restrict__ x,
                                  const float* __restrict__ W_bl,
                                  const float* __restrict__ b_bl,
                                  const float* __restrict__ W_off,
                                  const float* __restrict__ b_off,
                                  const float* __restrict__ W_def,
                                  const float* __restrict__ b_def,
                                  float* __restrict__ repres) {
  const int w = blockIdx.x * blockDim.x + threadIdx.x;
  const int h = blockIdx.y;
  const int b = blockIdx.z;
  if (w >= Wd) return;

  const float* xb = x + (size_t)b * Gg * HW;

  // ---- group_inner[k] = sum_g x[b,g,h,w] * W_bl[k,g] + b_bl[k]
  float gi[Gg];
#pragma unroll
  for (int k = 0; k < Gg; ++k) gi[k] = b_bl[k];
  for (int g = 0; g < Gg; ++g) {
    float xv = xb[g * HW + h * Wd + w];
#pragma unroll
    for (int k = 0; k < Gg; ++k) gi[k] += xv * W_bl[k * Gg + g];
  }

  // ---- mul_def accumulator over both dilation rates
  float md[Gg];
#pragma unroll
  for (int o = 0; o < Gg; ++o) md[o] = 0.f;

  for (int r = 0; r < NR; ++r) {
    const int rate = (r == 0) ? 1 : 2;

    // offset conv: 18 outputs, 3x3 dilated, same padding
    float off[18];
#pragma unroll
    for (int j = 0; j < 18; ++j) off[j] = b_off[r * 18 + j];
    for (int ky = 0; ky < 3; ++ky) {
      const int yy = h + rate * (ky - 1);
      if (yy < 0 || yy >= Hd) continue;
      for (int kx = 0; kx < 3; ++kx) {
        const int xx = w + rate * (kx - 1);
        if (xx < 0 || xx >= Wd) continue;
        const int t = ky * 3 + kx;
        for (int g = 0; g < Gg; ++g) {
          const float xv = xb[g * HW + yy * Wd + xx];
          const float* wo = W_off + (size_t)(r * 18 * Gg + g) * 9 + t;  // stride Gg*9 over j
#pragma unroll
          for (int j = 0; j < 18; ++j) off[j] += xv * wo[(size_t)j * Gg * 9];
        }
      }
    }

    // bilinear sample at 9 deformed positions + per-pixel matvec with W_def
    for (int kk = 0; kk < 9; ++kk) {
      const int ky = kk / 3, kx = kk % 3;
      const float py = off[2 * kk]     + (float)h + (float)(rate * (ky - 1));
      const float px = off[2 * kk + 1] + (float)w + (float)(rate * (kx - 1));
      const float y0f = floorf(py), x0f = floorf(px);
      const float wy = py - y0f, wx = px - x0f;
      const int y0 = (int)y0f, x0 = (int)x0f;

      const int   ys[2]  = { y0, y0 + 1 };
      const int   xs[2]  = { x0, x0 + 1 };
      const float wyv[2] = { 1.f - wy, wy };
      const float wxv[2] = { 1.f - wx, wx };

      for (int c = 0; c < Gg; ++c) {
        float s = 0.f;
#pragma unroll
        for (int iy = 0; iy < 2; ++iy) {
#pragma unroll
          for (int ix = 0; ix < 2; ++ix) {
            const int yi = ys[iy], xi = xs[ix];
            const bool ok = (yi >= 0) & (yi < Hd) & (xi >= 0) & (xi < Wd);
            const int yc = min(max(yi, 0), Hd - 1);
            const int xc = min(max(xi, 0), Wd - 1);
            const float v = xb[c * HW + yc * Wd + xc];
            s += ok ? v * wyv[iy] * wxv[ix] : 0.f;
          }
        }
        const float* wd = W_def + (size_t)(r * Gg * Gg + c) * 9 + kk;  // stride Gg*9 over o
#pragma unroll
        for (int o = 0; o < Gg; ++o) md[o] += s * wd[(size_t)o * Gg * 9];
      }
    }
#pragma unroll
    for (int o = 0; o < Gg; ++o) md[o] += b_def[r * Gg + o];
  }

  // ---- softmax over channel dim, gate, store
  float mx = md[0];
#pragma unroll
  for (int o = 1; o < Gg; ++o) mx = fmaxf(mx, md[o]);
  float sum = 0.f;
#pragma unroll
  for (int o = 0; o < Gg; ++o) { md[o] = __expf(md[o] - mx); sum += md[o]; }
  const float inv = 1.f / sum;
#pragma unroll
  for (int k = 0; k < Gg; ++k) {
    repres[((size_t)(b * Gg + k) * Hd + h) * Wd + w] = gi[k] * (1.f + md[k] * inv);
  }
}

// ---------------------------------------------------------------------------
// Kernel 2: ff = Conv2d(G,G,3,pad=1); one thread per pixel, all 30 outputs
// ---------------------------------------------------------------------------
__global__ void ff_conv_kernel(const float* __restrict__ rin,
                               const float* __restrict__ W_ff,
                               const float* __restrict__ b_ff,
                               float* __restrict__ rout) {
  const int w = blockIdx.x * blockDim.x + threadIdx.x;
  const int h = blockIdx.y;
  const int b = blockIdx.z;
  if (w >= Wd) return;

  float acc[Gg];
#pragma unroll
  for (int o = 0; o < Gg; ++o) acc[o] = b_ff[o];

  for (int ky = 0; ky < 3; ++ky) {
    const int yy = h + ky - 1;
    if (yy < 0 || yy >= Hd) continue;
    for (int kx = 0; kx < 3; ++kx) {
      const int xx = w + kx - 1;
      if (xx < 0 || xx >= Wd) continue;
      const int t = ky * 3 + kx;
      for (int c = 0; c < Gg; ++c) {
        const float v = rin[((size_t)(b * Gg + c) * Hd + yy) * Wd + xx];
        const float* wf = W_ff + (size_t)c * 9 + t;   // stride Gg*9 over o
#pragma unroll
        for (int o = 0; o < Gg; ++o) acc[o] += v * wf[(size_t)o * Gg * 9];
      }
    }
  }
#pragma unroll
  for (int o = 0; o < Gg; ++o)
    rout[((size_t)(b * Gg + o) * Hd + h) * Wd + w] = acc[o];
}

// ---------------------------------------------------------------------------
// Kernel 3: projection GEMM out[b,q,n] = sum_p Wp[q,p]*fea[b,p,n] + bp[q]
// FP32 WMMA 16x16x4. One wave computes a 48(M) x 32(N) register tile:
//   3 A-fragments x 2 B-fragments -> 6 WMMAs per K-step for
//   3 b64 (A) + 4 b32 (B) loads (~1.17 loads/WMMA vs 2.25 in the naive tile).
// ---------------------------------------------------------------------------
__global__ void proj_wmma_kernel(const float* __restrict__ fea,  // [B,720,862]
                                 const float* __restrict__ Wp,   // [720,720]
                                 const float* __restrict__ bp,   // [720]
                                 float* __restrict__ out) {      // [B,720,862]
  const int wave = (int)((blockIdx.x * blockDim.x + threadIdx.x) >> 5);
  const int lane = (int)(threadIdx.x & 31);
  const int tiles = Bb * MTW * NTW;
  if (wave >= tiles) return;

  const int b  = wave / (MTW * NTW);
  const int t  = wave % (MTW * NTW);
  const int mt = t / NTW, nt = t % NTW;
  const int q0 = mt * 48, n0 = nt * 32;

  const int half = lane >> 4;      // 0: K+{0,1}, 1: K+{2,3}
  const int lm   = lane & 15;
  const int kofs = half * 2;

  // A fragments: 3 row-panels of W_proj, one aligned b64 per panel per K-step
  const float* arow[3];
#pragma unroll
  for (int i = 0; i < 3; ++i)
    arow[i] = Wp + (size_t)(q0 + 16 * i + lm) * PREQ + kofs;

  // B fragments: 2 column-panels of fea, rows strided by Wd
  int  ncol[2];
  bool nok[2];
  const float* bcol[2];
#pragma unroll
  for (int j = 0; j < 2; ++j) {
    ncol[j] = n0 + 16 * j + lm;
    nok[j]  = (ncol[j] < Wd);
    bcol[j] = fea + ((size_t)b * PREQ + kofs) * Wd + (nok[j] ? ncol[j] : 0);
  }

  v8f acc[3][2] = {};

  for (int k = 0; k < PREQ; k += 16) {
    __builtin_prefetch(arow[0] + k + 64, 0, 1);   // global_prefetch_b8 on the A stream
#pragma unroll
    for (int kk = 0; kk < 16; kk += 4) {
      v2f a[3];
#pragma unroll
      for (int i = 0; i < 3; ++i)
        a[i] = *(const v2f*)(arow[i] + k + kk);

      v2f bf[2];
#pragma unroll
      for (int j = 0; j < 2; ++j) {
        const float t0 = bcol[j][(size_t)(k + kk)     * Wd];
        const float t1 = bcol[j][(size_t)(k + kk + 1) * Wd];
        bf[j].x = nok[j] ? t0 : 0.f;
        bf[j].y = nok[j] ? t1 : 0.f;
      }

#pragma unroll
      for (int i = 0; i < 3; ++i)
#pragma unroll
        for (int j = 0; j < 2; ++j)
          acc[i][j] = __builtin_amdgcn_wmma_f32_16x16x4_f32(
                          /*neg_a=*/false, a[i], /*neg_b=*/false, bf[j],
                          /*c_mod=*/(short)0, acc[i][j],
                          /*reuse_a=*/false, /*reuse_b=*/false);
    }
  }

  // C/D layout: VGPR e -> M = e + 8*half, N = lane&15
#pragma unroll
  for (int i = 0; i < 3; ++i) {
#pragma unroll
    for (int j = 0; j < 2; ++j) {
      if (!nok[j]) continue;
#pragma unroll
      for (int e = 0; e < 8; ++e) {
        const int m = q0 + 16 * i + e + half * 8;
        out[((size_t)b * PREQ + m) * Wd + ncol[j]] = acc[i][j][e] + bp[m];
      }
    }
  }
}

// ---------------------------------------------------------------------------
extern "C" void kernel_launch(void* const* d_in, const int* in_sizes, int n_in,
                              void* d_out, int out_size, void* d_ws, size_t ws_size,
                              hipStream_t stream) {
  const float* x      = (const float*)d_in[0];
  const float* W_bl   = (const float*)d_in[1];
  const float* b_bl   = (const float*)d_in[2];
  const float* W_off  = (const float*)d_in[3];
  const float* b_off  = (const float*)d_in[4];
  const float* W_def  = (const float*)d_in[5];
  const float* b_def  = (const float*)d_in[6];
  const float* W_ff   = (const float*)d_in[7];
  const float* b_ff   = (const float*)d_in[8];
  const float* W_proj = (const float*)d_in[9];
  const float* b_proj = (const float*)d_in[10];
  float* out = (float*)d_out;

  float* repres  = (float*)d_ws;                                  // [B,G,H,W]
  float* repres2 = repres + (size_t)Bb * Gg * Hd * Wd;            // [B,G,H,W] == [B,720,862]

  dim3 blk(128, 1, 1);
  dim3 grd((Wd + 127) / 128, Hd, Bb);
  fused_gate_kernel<<<grd, blk, 0, stream>>>(x, W_bl, b_bl, W_off, b_off,
                                             W_def, b_def, repres);
  ff_conv_kernel<<<grd, blk, 0, stream>>>(repres, W_ff, b_ff, repres2);

  const int tiles   = Bb * MTW * NTW;       // 6480 waves, 48x32 out each
  const int threads = 256;                  // 8 waves per block
  const int blocks  = (tiles * 32 + threads - 1) / threads;
  proj_wmma_kernel<<<blocks, threads, 0, stream>>>(repres2, W_proj, b_proj, out);
}